// our_PointsRasterizer_13855564497224
// MI455X (gfx1250) — compile-verified
//
#include <hip/hip_runtime.h>

#define HH 128
#define WW 128
#define KK 8
#define PP 8192
#define CHUNK 512
#define NCHUNK (PP / CHUNK)

typedef float v2f  __attribute__((ext_vector_type(2)));
typedef float v8f  __attribute__((ext_vector_type(8)));
typedef unsigned int u32x4 __attribute__((ext_vector_type(4)));
typedef int   i32x4 __attribute__((ext_vector_type(4)));
typedef int   i32x8 __attribute__((ext_vector_type(8)));

struct ChunkBuf {
  float pts[CHUNK * 3];   // x,y,z interleaved
  float rad[CHUNK];       // per-point radius
};

// 1-D dword copy via the Tensor Data Mover. D# per CDNA5 ISA:
// group0: [1:0]=count=1, [63:32]=lds_addr, [120:64]=global_addr, [127:126]=type 2
// group1: data_size=4B (code 2), tensor_dim0=tile_dim0=ndw, tensor_dim1=1, tile_dim1=0 (unused)
__device__ __forceinline__ void tdm_copy_dwords(unsigned long long gaddr,
                                                unsigned ldsByte, unsigned ndw) {
  u32x4 g0;
  g0[0] = 1u;
  g0[1] = ldsByte;
  g0[2] = (unsigned)(gaddr & 0xFFFFFFFFull);
  g0[3] = (unsigned)((gaddr >> 32) & 0x01FFFFFFull) | (2u << 30);
  i32x8 g1;
  g1[0] = (int)(2u << 16);                 // workgroup_mask=0, data_size=4B
  g1[1] = (int)((ndw & 0xFFFFu) << 16);    // tensor_dim0[15:0]
  g1[2] = (int)((ndw >> 16) | (1u << 16)); // tensor_dim0[31:16] | tensor_dim1=1
  g1[3] = (int)((ndw & 0xFFFFu) << 16);    // tensor_dim1[31:16]=0 | tile_dim0
  g1[4] = 0;                               // tile_dim1=0 (unused), tile_dim2=0
  g1[5] = (int)ndw;                        // tensor_dim0_stride lo32 (unused for 1 row)
  g1[6] = 0;
  g1[7] = 0;
  i32x4 z4 = {0, 0, 0, 0};
  i32x8 z8 = {0, 0, 0, 0, 0, 0, 0, 0};
  // 6-arg toolchain variant: (g0, g1, g2, g3, extra, cpol)
  __builtin_amdgcn_tensor_load_to_lds(g0, g1, z4, z4, z8, 0);
}

#define CSWAP(a, b)                                                          \
  do {                                                                       \
    const bool sw_ = (mz[b] < mz[a]) || ((mz[b] == mz[a]) && (mi[b] < mi[a])); \
    const float tz_ = sw_ ? mz[b] : mz[a];                                   \
    mz[b] = sw_ ? mz[a] : mz[b]; mz[a] = tz_;                                \
    const float td_ = sw_ ? md[b] : md[a];                                   \
    md[b] = sw_ ? md[a] : md[b]; md[a] = td_;                                \
    const int ti_ = sw_ ? mi[b] : mi[a];                                     \
    mi[b] = sw_ ? mi[a] : mi[b]; mi[a] = ti_;                                \
  } while (0)

__global__ __launch_bounds__(256, 1) void our_rasterize_topk_wmma(
    const float* __restrict__ points, const float* __restrict__ radius,
    float* __restrict__ out) {
  __shared__ ChunkBuf sbuf[2];

  const int tid  = (int)threadIdx.x;
  const int lane = tid & 31;
  const int n    = lane & 15;     // pixel column within wave tile; also A-row m
  const bool hi  = lane >= 16;    // lane half selects K rows {2,3} and M offset +8
  const int wv   = tid >> 5;      // wave id: 8 waves = 8 pixel groups of 16
  const int h    = (int)blockIdx.x;
  const int w    = wv * 16 + n;

  const float xf = 1.0f - (2.0f * (float)w + 1.0f) * (1.0f / (float)WW);
  const float yf = 1.0f - (2.0f * (float)h + 1.0f) * (1.0f / (float)HH);

  // B fragment (4x16, pixels): col n = [xf, yf, 1, xf^2+yf^2]
  v2f bfrag;
  bfrag.x = hi ? 1.0f : xf;                 // K=2 : K=0
  bfrag.y = hi ? (xf * xf + yf * yf) : yf;  // K=3 : K=1

  float kz[KK], kd[KK];
  int ki[KK];
#pragma unroll
  for (int j = 0; j < KK; ++j) { kz[j] = __builtin_inff(); kd[j] = -1.0f; ki[j] = -1; }

  const unsigned lds0 = (unsigned)(unsigned long long)(&sbuf[0]);
  const unsigned lds1 = (unsigned)(unsigned long long)(&sbuf[1]);
  const unsigned radOff = (unsigned)(CHUNK * 3 * sizeof(float));

  // Prologue: wave 0 kicks off DMA of chunk 0.
  if (tid < 32) {
    tdm_copy_dwords((unsigned long long)points, lds0, CHUNK * 3);
    tdm_copy_dwords((unsigned long long)radius, lds0 + radOff, CHUNK);
  }

  for (int c = 0; c < NCHUNK; ++c) {
    if (tid < 32) __builtin_amdgcn_s_wait_tensorcnt(0);
    __syncthreads();  // chunk c resident; prior consumers of the other buffer done
    if (((c + 1) < NCHUNK) && (tid < 32)) {
      const unsigned dst = ((c + 1) & 1) ? lds1 : lds0;
      const unsigned long long pofs = (unsigned long long)(c + 1) * CHUNK;
      tdm_copy_dwords((unsigned long long)(points + pofs * 3), dst, CHUNK * 3);
      tdm_copy_dwords((unsigned long long)(radius + pofs), dst + radOff, CHUNK);
    }
    const ChunkBuf* B = &sbuf[c & 1];
    const int pbase = c * CHUNK;

#pragma unroll 2
    for (int t = 0; t < CHUNK; t += 16) {
      // A fragment (16x4, points): row m = [-2px, -2py, px^2+py^2, 1]
      const float px = B->pts[3 * (t + n) + 0];
      const float py = B->pts[3 * (t + n) + 1];
      v2f afrag;
      afrag.x = hi ? (px * px + py * py) : (-2.0f * px);  // K=2 : K=0
      afrag.y = hi ? 1.0f : (-2.0f * py);                 // K=3 : K=1

      v8f acc = {0.f, 0.f, 0.f, 0.f, 0.f, 0.f, 0.f, 0.f};
      acc = __builtin_amdgcn_wmma_f32_16x16x4_f32(false, afrag, false, bfrag,
                                                  (short)0, acc, false, false);

      // D layout: VGPR r holds dist2 of point (t + r + hi*8) vs this lane's pixel.
      const int mb = t + (hi ? 8 : 0);
#pragma unroll
      for (int r = 0; r < 8; ++r) {
        const float z  = B->pts[3 * (mb + r) + 2];
        const float rr = B->rad[mb + r];
        const float d2 = acc[r];
        const bool ok = (d2 < rr * rr) && (z >= 0.0f) && (z < kz[KK - 1]);
        if (ok) {
          float cz = z, cd = d2;
          int ci = pbase + mb + r;
#pragma unroll
          for (int j = 0; j < KK; ++j) {  // sorted insert, push largest out
            const bool sm = cz < kz[j];
            const float tz = sm ? kz[j] : cz;
            const float td = sm ? kd[j] : cd;
            const int ti = sm ? ki[j] : ci;
            kz[j] = sm ? cz : kz[j];
            kd[j] = sm ? cd : kd[j];
            ki[j] = sm ? ci : ki[j];
            cz = tz; cd = td; ci = ti;
          }
        }
      }
    }
    __syncthreads();  // done reading buf[c&1] before it gets overwritten
  }

  // Merge this lane's sorted top-8 with partner lane (same pixel, other point half).
  float pz[KK], pd[KK];
  int pi[KK];
#pragma unroll
  for (int j = 0; j < KK; ++j) {
    pz[j] = __shfl_xor(kz[j], 16, 32);
    pd[j] = __shfl_xor(kd[j], 16, 32);
    pi[j] = __shfl_xor(ki[j], 16, 32);
  }
  // Bitonic lower half: 8 smallest of (A asc ++ rev(B asc)), key = (z, idx)
  float mz[KK], md[KK];
  int mi[KK];
#pragma unroll
  for (int j = 0; j < KK; ++j) {
    const float az = kz[j], bz = pz[KK - 1 - j];
    const float ad = kd[j], bd = pd[KK - 1 - j];
    const int ai = ki[j], bi = pi[KK - 1 - j];
    const bool tA = (az < bz) || ((az == bz) && (ai <= bi));
    mz[j] = tA ? az : bz;
    md[j] = tA ? ad : bd;
    mi[j] = tA ? ai : bi;
  }
  // Bitonic clean (sort the bitonic 8-sequence ascending).
  CSWAP(0, 4); CSWAP(1, 5); CSWAP(2, 6); CSWAP(3, 7);
  CSWAP(0, 2); CSWAP(1, 3); CSWAP(4, 6); CSWAP(5, 7);
  CSWAP(0, 1); CSWAP(2, 3); CSWAP(4, 5); CSWAP(6, 7);

  if (!hi) {  // both halves hold identical results; low half writes
    const int pix = h * WW + w;
    const int HWK = HH * WW * KK;
#pragma unroll
    for (int j = 0; j < KK; ++j) {
      const bool v = mi[j] >= 0;
      out[pix * KK + j]            = (float)mi[j];          // idx plane (-1 pad)
      out[HWK + pix * KK + j]      = v ? mz[j] : -1.0f;     // zbuf plane
      out[2 * HWK + pix * KK + j]  = md[j];                 // dists plane (-1 pad)
    }
  }
}

extern "C" void kernel_launch(void* const* d_in, const int* in_sizes, int n_in,
                              void* d_out, int out_size, void* d_ws, size_t ws_size,
                              hipStream_t stream) {
  (void)in_sizes; (void)n_in; (void)d_ws; (void)ws_size; (void)out_size;
  const float* points = (const float*)d_in[0];
  const float* radius = (const float*)d_in[1];
  float* out = (float*)d_out;
  // One block per image row: 8 wave32 x 16 pixels = 128 pixels.
  our_rasterize_topk_wmma<<<dim3(HH), dim3(256), 0, stream>>>(points, radius, out);
}